// CosineDistance_8581344658087
// MI455X (gfx1250) — compile-verified
//
#include <hip/hip_runtime.h>

#ifndef __has_builtin
#define __has_builtin(x) 0
#endif

#define S_DIM 512
#define B_DIM 4096
#define D_DIM 64
#define EPSF  1e-10f

#define WAVES        8
#define BLOCK        (WAVES * 32)
#define B_PER_WAVE   32
#define B_PER_BLOCK  (WAVES * B_PER_WAVE)   // 256
#define S_CHUNK      16
#define ROWF         68                      // padded row stride in floats (272 B)
#define TILEF        (B_PER_WAVE * ROWF)     // 2176 floats = 8704 B per buffer

// 4 x i32 vector type matching the builtin's expected pointee
typedef int v4i __attribute__((vector_size(16)));
typedef __attribute__((address_space(1))) v4i* global_v4i_p;
typedef __attribute__((address_space(3))) v4i* lds_v4i_p;

// ---- CDNA5 async global->LDS copy (ASYNCcnt path) -------------------------
__device__ __forceinline__ void async_b128(const float* g, float* l) {
#if __has_builtin(__builtin_amdgcn_global_load_async_to_lds_b128)
    __builtin_amdgcn_global_load_async_to_lds_b128(
        (global_v4i_p)(v4i*)(void*)(float*)g,
        (lds_v4i_p)(v4i*)(void*)l,
        /*offset=*/0, /*cpol=*/0);
#else
    unsigned lds_addr = (unsigned)(unsigned long long)l;   // low 32 bits = LDS offset
    unsigned long long gaddr = (unsigned long long)g;
    asm volatile("global_load_async_to_lds_b128 %0, %1, off"
                 :: "v"(lds_addr), "v"(gaddr)
                 : "memory");
#endif
}

template <int N>
__device__ __forceinline__ void wait_asynccnt() {
#if __has_builtin(__builtin_amdgcn_s_wait_asynccnt)
    __builtin_amdgcn_s_wait_asynccnt(N);
#else
    if (N == 0) asm volatile("s_wait_asynccnt 0"  ::: "memory");
    else        asm volatile("s_wait_asynccnt 16" ::: "memory");
#endif
    asm volatile("" ::: "memory");   // compile-time ordering fence
}

// ---------------------------------------------------------------------------
__global__ __launch_bounds__(BLOCK) void CosineDistance_kernel(
    const float* __restrict__ support,   // [S, B, D]
    const float* __restrict__ xhat,      // [B, D]
    const int*   __restrict__ normp,     // [1]
    float*       __restrict__ out)       // [S, B]
{
    __shared__ __align__(16) float smem[WAVES * 2 * TILEF];   // ~136 KB

    const int lane = threadIdx.x & 31;
    const int wave = threadIdx.x >> 5;
    const int b0   = blockIdx.x * B_PER_BLOCK + wave * B_PER_WAVE;
    const int b    = b0 + lane;
    const int s0   = blockIdx.y * S_CHUNK;
    const int norm = *normp;

    float* wbase = &smem[wave * 2 * TILEF];

    // ---- X_hat row for this lane: 64 floats kept in VGPRs, read once ------
    float4 xh[16];
    const float4* xp = reinterpret_cast<const float4*>(xhat + (size_t)b * D_DIM);
    float nx = 0.0f;
    #pragma unroll
    for (int k = 0; k < 16; ++k) {
        xh[k] = xp[k];
        nx = fmaf(xh[k].x, xh[k].x, nx);
        nx = fmaf(xh[k].y, xh[k].y, nx);
        nx = fmaf(xh[k].z, xh[k].z, nx);
        nx = fmaf(xh[k].w, xh[k].w, nx);
    }
    const float inv_xn = 1.0f / fmaxf(sqrtf(nx), EPSF);

    // per-lane addressing for the coalesced async copy:
    // chunk c = i*32 + lane (16 B each) -> row r = c>>4, 16B-slot k = c&15
    const int ldsBase = (lane >> 4) * ROWF + (lane & 15) * 4;   // floats

    // issue one 8 KB tile support[s, b0:b0+32, :] into buffer `buf`
    auto issue_tile = [&](int s, int buf) {
        const float* g = support + ((size_t)s * B_DIM + (size_t)b0) * D_DIM + lane * 4;
        float*       l = wbase + buf * TILEF + ldsBase;
        #pragma unroll
        for (int i = 0; i < 16; ++i) {
            async_b128(g, l);      // 512 B per instruction, fully coalesced
            g += 128;              // 32 chunks * 4 floats
            l += 2 * ROWF;         // 2 padded rows
        }
    };

    issue_tile(s0, 0);

    for (int t = 0; t < S_CHUNK; ++t) {
        const int s = s0 + t;
        if (t + 1 < S_CHUNK) {
            issue_tile(s + 1, (t + 1) & 1);   // prefetch next tile
            wait_asynccnt<16>();              // in-order: current 16 are done
        } else {
            wait_asynccnt<0>();
        }

        // lane reads its own padded row: bank-conflict-free ds_load_b128
        const float4* row = reinterpret_cast<const float4*>(
            wbase + (t & 1) * TILEF + lane * ROWF);

        float dot = 0.0f, ns = 0.0f;
        #pragma unroll
        for (int k = 0; k < 16; ++k) {
            float4 v = row[k];
            float4 x = xh[k];
            dot = fmaf(v.x, x.x, dot);
            dot = fmaf(v.y, x.y, dot);
            dot = fmaf(v.z, x.z, dot);
            dot = fmaf(v.w, x.w, dot);
            ns  = fmaf(v.x, v.x, ns);
            ns  = fmaf(v.y, v.y, ns);
            ns  = fmaf(v.z, v.z, ns);
            ns  = fmaf(v.w, v.w, ns);
        }

        float sim = (dot / fmaxf(sqrtf(ns), EPSF)) * inv_xn;
        if (norm) sim = (sim + 1.0f) * 0.5f;
        out[(size_t)s * B_DIM + b] = sim;
    }
}

// ---------------------------------------------------------------------------
extern "C" void kernel_launch(void* const* d_in, const int* in_sizes, int n_in,
                              void* d_out, int out_size, void* d_ws, size_t ws_size,
                              hipStream_t stream) {
    const float* support = (const float*)d_in[0];   // [512, 4096, 64] f32
    const float* xhat    = (const float*)d_in[1];   // [4096, 64] f32
    const int*   normp   = (const int*)d_in[2];     // scalar int
    float*       out     = (float*)d_out;           // [512, 4096] f32

    dim3 grid(B_DIM / B_PER_BLOCK, S_DIM / S_CHUNK);   // 16 x 32 = 512 WGs
    dim3 block(BLOCK);
    CosineDistance_kernel<<<grid, block, 0, stream>>>(support, xhat, normp, out);
}